// SFLBlock_43920335569212
// MI455X (gfx1250) — compile-verified
//
#include <hip/hip_runtime.h>
#include <hip/hip_bf16.h>
#include <math.h>

// =====================================================================
// SFLBlock forward for MI455X (gfx1250, wave32, WMMA).
// All GEMMs use V_WMMA_F32_16X16X4_F32 (fp32 in/out, full precision).
// Register blocking: each wave computes a 16x80 output strip (5 tiles),
// reusing the A fragment across the 5 column tiles per K-step.
// Scratch requirement: ~299 MB in d_ws (assumed available).
// =====================================================================

#define Bd 8
#define Cd 128
#define Nd 2000
#define Kn 9
#define Ed 32
#define NK 18000           // Nd*Kn
#define EPS_CN 1e-3f
#define EPS_AF 1e-5f
#define BNS 0.9999950000374996f   // 1/sqrt(1+1e-5)

typedef __attribute__((ext_vector_type(2))) float v2f;
typedef __attribute__((ext_vector_type(8))) float v8f;

__device__ __forceinline__ v8f wmma4(v2f a, v2f b, v8f c) {
  // D(16x16,f32) = A(16x4,f32) * B(4x16,f32) + C
  return __builtin_amdgcn_wmma_f32_16x16x4_f32(
      /*neg_a=*/false, a, /*neg_b=*/false, b,
      /*c_mod=*/(short)0, c, /*reuse_a=*/false, /*reuse_b=*/false);
}

union AccU { v8f v; float f[8]; };

// ---------------------------------------------------------------------
// 0a. xx[b,n] = sum_c x^2
// ---------------------------------------------------------------------
__global__ void xx_kernel(const float* __restrict__ x, float* __restrict__ xx) {
  int t = blockIdx.x * blockDim.x + threadIdx.x;
  if (t >= Bd * Nd) return;
  int b = t / Nd, n = t % Nd;
  const float* xp = x + (size_t)b * Cd * Nd + n;
  float s = 0.f;
  for (int c = 0; c < Cd; c++) { float v = xp[(size_t)c * Nd]; s += v * v; }
  xx[t] = s;
}

// 0b. xT[b,n,c] = x[b,c,n]  (makes Gram fragment loads contiguous)
__global__ void transpose_kernel(const float* __restrict__ x, float* __restrict__ xT) {
  int t = blockIdx.x * blockDim.x + threadIdx.x;
  if (t >= Bd * Cd * Nd) return;
  int n = t % Nd;
  int bc = t / Nd;
  int b = bc / Cd, c = bc % Cd;
  xT[((size_t)b * Nd + n) * Cd + c] = x[t];
}

// ---------------------------------------------------------------------
// 1. KNN distance: d[b,n,m] = -xx[n] + 2*(x^T x)[n,m] - xx[m]
//    Both operands from xT (b,n,c): contiguous pair loads.
//    Wave computes 16 rows x 80 cols. grid (25,125,8), block 32
// ---------------------------------------------------------------------
__global__ void knn_dist_kernel(const float* __restrict__ xT,
                                const float* __restrict__ xx,
                                float* __restrict__ G) {
  const int NB = 5;
  int nt = blockIdx.x, mt = blockIdx.y, b = blockIdx.z;
  int lane = threadIdx.x, l15 = lane & 15, hk = lane >> 4;
  const float* xb = xT + (size_t)b * Nd * Cd;
  int rowA = mt * 16 + l15;
  int col0 = nt * 16 * NB + l15;
  v8f acc[NB];
#pragma unroll
  for (int t = 0; t < NB; t++) acc[t] = (v8f){};
  for (int k = 0; k < Cd; k += 4) {
    int k0 = k + hk * 2;
    v2f a;
    a.x = xb[(size_t)rowA * Cd + k0];
    a.y = xb[(size_t)rowA * Cd + k0 + 1];
#pragma unroll
    for (int t = 0; t < NB; t++) {
      int colB = col0 + t * 16;
      v2f bb;
      bb.x = xb[(size_t)colB * Cd + k0];
      bb.y = xb[(size_t)colB * Cd + k0 + 1];
      acc[t] = wmma4(a, bb, acc[t]);
    }
  }
  float* Gb = G + (size_t)b * Nd * Nd;
  int rowbase = mt * 16 + hk * 8;
#pragma unroll
  for (int t = 0; t < NB; t++) {
    AccU u; u.v = acc[t];
    int colB = col0 + t * 16;
    float xc = xx[b * Nd + colB];
    for (int r = 0; r < 8; r++) {
      int row = rowbase + r;
      Gb[(size_t)row * Nd + colB] = 2.f * u.f[r] - xx[b * Nd + row] - xc;
    }
  }
}

// ---------------------------------------------------------------------
// 2. top-9 per row (descending, ties -> lower index first)
// ---------------------------------------------------------------------
__global__ void topk_kernel(const float* __restrict__ G, int* __restrict__ idx) {
  int t = blockIdx.x * blockDim.x + threadIdx.x;
  if (t >= Bd * Nd) return;
  int b = t / Nd, row = t % Nd;
  const float* g = G + (size_t)b * Nd * Nd + (size_t)row * Nd;
  float bv[Kn]; int bi[Kn];
  for (int i = 0; i < Kn; i++) { bv[i] = -3.4e38f; bi[i] = 0; }
  for (int m = 0; m < Nd; m++) {
    float v = g[m];
    if (v > bv[Kn - 1]) {
      int p = Kn - 1;
      while (p > 0 && bv[p - 1] < v) { bv[p] = bv[p - 1]; bi[p] = bi[p - 1]; p--; }
      bv[p] = v; bi[p] = m;
    }
  }
  for (int i = 0; i < Kn; i++) idx[t * Kn + i] = bi[i];
}

// ---------------------------------------------------------------------
// 3. att1 conv over fused graph feature (gather fused into B fragments):
//    hpre[b,co,j] = sum_ci W[co,ci]*xl[ci,j] + bias, xl = [center;center-neigh]
//    Wave: 16 rows x 80 j-cols. grid (225, 8, 8), block 32
// ---------------------------------------------------------------------
__global__ void local_h_kernel(const float* __restrict__ x, const int* __restrict__ idx,
                               const float* __restrict__ W, const float* __restrict__ bias,
                               float* __restrict__ hpre) {
  const int NB = 5;
  int jt = blockIdx.x, mt = blockIdx.y, b = blockIdx.z;
  int lane = threadIdx.x, l15 = lane & 15, hk = lane >> 4;
  const float* xb = x + (size_t)b * Cd * Nd;
  int rowA = mt * 16 + l15;
  int nn[NB], nbr[NB];
#pragma unroll
  for (int t = 0; t < NB; t++) {
    int col = jt * 16 * NB + t * 16 + l15;
    int n = col / Kn, kn = col - n * Kn;
    nn[t] = n;
    nbr[t] = idx[(b * Nd + n) * Kn + kn];
  }
  v8f acc[NB];
#pragma unroll
  for (int t = 0; t < NB; t++) acc[t] = (v8f){};
  for (int k = 0; k < 2 * Cd; k += 4) {
    int k0 = k + hk * 2;
    v2f a;
    a.x = W[rowA * (2 * Cd) + k0];
    a.y = W[rowA * (2 * Cd) + k0 + 1];
    int c0 = k0, c1 = k0 + 1;
#pragma unroll
    for (int t = 0; t < NB; t++) {
      v2f bb;
      bb.x = (c0 < Cd) ? xb[(size_t)c0 * Nd + nn[t]]
                       : (xb[(size_t)(c0 - Cd) * Nd + nn[t]] - xb[(size_t)(c0 - Cd) * Nd + nbr[t]]);
      bb.y = (c1 < Cd) ? xb[(size_t)c1 * Nd + nn[t]]
                       : (xb[(size_t)(c1 - Cd) * Nd + nn[t]] - xb[(size_t)(c1 - Cd) * Nd + nbr[t]]);
      acc[t] = wmma4(a, bb, acc[t]);
    }
  }
  float* hp = hpre + (size_t)b * Cd * NK;
  int rowbase = mt * 16 + hk * 8;
#pragma unroll
  for (int t = 0; t < NB; t++) {
    AccU u; u.v = acc[t];
    int col = jt * 16 * NB + t * 16 + l15;
    for (int r = 0; r < 8; r++) {
      int row = rowbase + r;
      hp[(size_t)row * NK + col] = u.f[r] + bias[row];
    }
  }
}

// ---------------------------------------------------------------------
// 4. per-(b,c) mean / rstd over S spatial elements (instance norm stats)
// ---------------------------------------------------------------------
__global__ void stats_kernel(const float* __restrict__ src, float* __restrict__ mbuf,
                             float* __restrict__ ibuf, int S, float eps) {
  int ch = blockIdx.x;
  const float* p = src + (size_t)ch * S;
  float s = 0.f, s2 = 0.f;
  for (int i = threadIdx.x; i < S; i += 256) { float v = p[i]; s += v; s2 += v * v; }
  __shared__ float sh[256], sh2[256];
  sh[threadIdx.x] = s; sh2[threadIdx.x] = s2; __syncthreads();
  for (int o = 128; o > 0; o >>= 1) {
    if (threadIdx.x < (unsigned)o) { sh[threadIdx.x] += sh[threadIdx.x + o]; sh2[threadIdx.x] += sh2[threadIdx.x + o]; }
    __syncthreads();
  }
  if (threadIdx.x == 0) {
    float m = sh[0] / S;
    float v = sh2[0] / S - m * m;
    mbuf[ch] = m;
    ibuf[ch] = rsqrtf(v + eps);
  }
}

// ---------------------------------------------------------------------
// 5. fused local transformer per point (LDS-staged H, 3 WMMA GEMMs,
//    softmax over k=9, weighted sum, gamma-residual)
// grid (Nd, Bd), block 256 (8 waves)
// ---------------------------------------------------------------------
__global__ __launch_bounds__(256) void local_attn_kernel(
    const float* __restrict__ x, const float* __restrict__ hpre,
    const float* __restrict__ mbuf, const float* __restrict__ ibuf,
    const float* __restrict__ g_in, const float* __restrict__ b_in,
    const float* __restrict__ Wq, const float* __restrict__ bq,
    const float* __restrict__ Wk, const float* __restrict__ bk,
    const float* __restrict__ Wv, const float* __restrict__ bv,
    const float* __restrict__ gamma, float* __restrict__ out) {
  int n = blockIdx.x, b = blockIdx.y;
  __shared__ float H[Cd][16];
  __shared__ float QK[8][16][16];
  __shared__ float VV[8][16][16];
  const float* hp = hpre + (size_t)b * Cd * NK;
  for (int t = threadIdx.x; t < Cd * 16; t += 256) {
    int ci = t >> 4, kn = t & 15;
    float v = 0.f;
    if (kn < Kn) {
      float raw = hp[(size_t)ci * NK + n * Kn + kn];
      float m = mbuf[b * Cd + ci], iv = ibuf[b * Cd + ci];
      v = fmaxf(g_in[ci] * (raw - m) * iv * BNS + b_in[ci], 0.f);
    }
    H[ci][kn] = v;
  }
  __syncthreads();
  int wave = threadIdx.x >> 5, lane = threadIdx.x & 31;
  int l15 = lane & 15, hk = lane >> 4;
  int rowA = wave * 16 + l15;
  v8f aq = {}, ak = {}, av = {};
  for (int k = 0; k < Cd; k += 4) {
    int k0 = k + hk * 2;
    v2f bfrag; bfrag.x = H[k0][l15]; bfrag.y = H[k0 + 1][l15];
    v2f a;
    a.x = Wq[rowA * Cd + k0]; a.y = Wq[rowA * Cd + k0 + 1];
    aq = wmma4(a, bfrag, aq);
    a.x = Wk[rowA * Cd + k0]; a.y = Wk[rowA * Cd + k0 + 1];
    ak = wmma4(a, bfrag, ak);
    a.x = Wv[rowA * Cd + k0]; a.y = Wv[rowA * Cd + k0 + 1];
    av = wmma4(a, bfrag, av);
  }
  AccU uq, uk, uv; uq.v = aq; uk.v = ak; uv.v = av;
  int rowbase = hk * 8;
  for (int r = 0; r < 8; r++) {
    int row = rowbase + r;
    int ch = wave * 16 + row;
    float q = uq.f[r] + bq[ch];
    float kk = uk.f[r] + bk[ch];
    QK[wave][row][l15] = q * kk;
    VV[wave][row][l15] = uv.f[r] + bv[ch];
  }
  __syncthreads();
  if (lane < 16) {
    int ch = wave * 16 + lane;
    float mx = -3.4e38f;
    for (int j = 0; j < Kn; j++) mx = fmaxf(mx, QK[wave][lane][j]);
    float se = 0.f, o = 0.f;
    for (int j = 0; j < Kn; j++) {
      float e = __expf(QK[wave][lane][j] - mx);
      se += e;
      o += e * VV[wave][lane][j];
    }
    o /= se;
    size_t oi = ((size_t)b * Cd + ch) * Nd + n;
    out[oi] = x[oi] + gamma[0] * o;
  }
}

// ---------------------------------------------------------------------
// 6. generic 1x1-conv GEMM, 16x(16*NB) strip per wave, fused pre/epi.
//    PRE: 0 identity, 1 relu(g*(x-m)*inv*BNS + bb)
//    EPI: 0 plain, 1 relu(g*o*BNS+bb), 2 +resid, 3 resid + gamma*relu(...)
// grid (S/(16*NB), M/16, Bd), block 32
// ---------------------------------------------------------------------
template <int PRE, int EPI, int NB>
__global__ void gemm_kernel(const float* __restrict__ W, const float* __restrict__ bias,
                            const float* __restrict__ Xin, int Kdim, int S,
                            const float* __restrict__ m_in, const float* __restrict__ i_in,
                            const float* __restrict__ g_in, const float* __restrict__ b_in,
                            const float* __restrict__ g_out, const float* __restrict__ b_out,
                            const float* __restrict__ resid, const float* __restrict__ gamma,
                            float* __restrict__ out) {
  int nt = blockIdx.x, mt = blockIdx.y, b = blockIdx.z;
  int lane = threadIdx.x, l15 = lane & 15, hk = lane >> 4;
  const float* Xb = Xin + (size_t)b * Kdim * S;
  int col0 = nt * 16 * NB + l15;
  int rowA = mt * 16 + l15;
  int Mdim = gridDim.y * 16;
  if (EPI == 2 || EPI == 3) {
    // warm the residual rows while the MAC loop runs (global_prefetch_b8)
    __builtin_prefetch(&resid[(size_t)b * Mdim * S + (size_t)(mt * 16) * S + col0], 0, 1);
  }
  v8f acc[NB];
#pragma unroll
  for (int t = 0; t < NB; t++) acc[t] = (v8f){};
  for (int k = 0; k < Kdim; k += 4) {
    int k0 = k + hk * 2;
    v2f a;
    a.x = W[rowA * Kdim + k0];
    a.y = W[rowA * Kdim + k0 + 1];
    float gm0, gm1, ad0, ad1;
    if (PRE == 1) {
      int c0 = b * Kdim + k0;
      gm0 = g_in[k0] * i_in[c0] * BNS;
      gm1 = g_in[k0 + 1] * i_in[c0 + 1] * BNS;
      ad0 = b_in[k0] - gm0 * m_in[c0];
      ad1 = b_in[k0 + 1] - gm1 * m_in[c0 + 1];
    }
#pragma unroll
    for (int t = 0; t < NB; t++) {
      int col = col0 + t * 16;
      float v0 = Xb[(size_t)k0 * S + col];
      float v1 = Xb[(size_t)(k0 + 1) * S + col];
      if (PRE == 1) {
        v0 = fmaxf(gm0 * v0 + ad0, 0.f);
        v1 = fmaxf(gm1 * v1 + ad1, 0.f);
      }
      v2f bb; bb.x = v0; bb.y = v1;
      acc[t] = wmma4(a, bb, acc[t]);
    }
  }
  float* ob = out + (size_t)b * Mdim * S;
  int rowbase = mt * 16 + hk * 8;
#pragma unroll
  for (int t = 0; t < NB; t++) {
    AccU u; u.v = acc[t];
    int col = col0 + t * 16;
    for (int r = 0; r < 8; r++) {
      int row = rowbase + r;
      float v = u.f[r] + bias[row];
      if (EPI == 1) v = fmaxf(g_out[row] * v * BNS + b_out[row], 0.f);
      if (EPI == 2) v += resid[(size_t)b * Mdim * S + (size_t)row * S + col];
      if (EPI == 3) {
        v = fmaxf(g_out[row] * v * BNS + b_out[row], 0.f);
        v = resid[(size_t)b * Mdim * S + (size_t)row * S + col] + gamma[0] * v;
      }
      ob[(size_t)row * S + col] = v;
    }
  }
}

// ---------------------------------------------------------------------
// 7. scores[b,n,m] = sum_c q1[b,c,n]*k1[b,c,m]  (K=32)
//    Wave: 16 rows x 80 cols. grid (25,125,8), block 32
// ---------------------------------------------------------------------
__global__ void scores_kernel(const float* __restrict__ q1, const float* __restrict__ k1,
                              float* __restrict__ G) {
  const int NB = 5;
  int nt = blockIdx.x, mt = blockIdx.y, b = blockIdx.z;
  int lane = threadIdx.x, l15 = lane & 15, hk = lane >> 4;
  const float* qb = q1 + (size_t)b * Ed * Nd;
  const float* kb = k1 + (size_t)b * Ed * Nd;
  int rowA = mt * 16 + l15;
  int col0 = nt * 16 * NB + l15;
  v8f acc[NB];
#pragma unroll
  for (int t = 0; t < NB; t++) acc[t] = (v8f){};
  for (int k = 0; k < Ed; k += 4) {
    int k0 = k + hk * 2;
    v2f a;
    a.x = qb[(size_t)k0 * Nd + rowA];
    a.y = qb[(size_t)(k0 + 1) * Nd + rowA];
#pragma unroll
    for (int t = 0; t < NB; t++) {
      int colB = col0 + t * 16;
      v2f bb;
      bb.x = kb[(size_t)k0 * Nd + colB];
      bb.y = kb[(size_t)(k0 + 1) * Nd + colB];
      acc[t] = wmma4(a, bb, acc[t]);
    }
  }
  float* Gb = G + (size_t)b * Nd * Nd;
  int rowbase = mt * 16 + hk * 8;
#pragma unroll
  for (int t = 0; t < NB; t++) {
    AccU u; u.v = acc[t];
    int colB = col0 + t * 16;
    for (int r = 0; r < 8; r++) Gb[(size_t)(rowbase + r) * Nd + colB] = u.f[r];
  }
}

// ---------------------------------------------------------------------
// 8. in-place row softmax over 2000 cols, grid (Bd*Nd), block 128
// ---------------------------------------------------------------------
__global__ void softmax_kernel(float* __restrict__ G) {
  int t = blockIdx.x;
  int b = t / Nd, row = t % Nd;
  float* g = G + (size_t)b * Nd * Nd + (size_t)row * Nd;
  __shared__ float sh[128];
  float mx = -3.4e38f;
  for (int i = threadIdx.x; i < Nd; i += 128) mx = fmaxf(mx, g[i]);
  sh[threadIdx.x] = mx; __syncthreads();
  for (int o = 64; o > 0; o >>= 1) {
    if (threadIdx.x < (unsigned)o) sh[threadIdx.x] = fmaxf(sh[threadIdx.x], sh[threadIdx.x + o]);
    __syncthreads();
  }
  mx = sh[0]; __syncthreads();
  float s = 0.f;
  for (int i = threadIdx.x; i < Nd; i += 128) { float e = __expf(g[i] - mx); g[i] = e; s += e; }
  sh[threadIdx.x] = s; __syncthreads();
  for (int o = 64; o > 0; o >>= 1) {
    if (threadIdx.x < (unsigned)o) sh[threadIdx.x] += sh[threadIdx.x + o];
    __syncthreads();
  }
  float inv = 1.f / sh[0];
  for (int i = threadIdx.x; i < Nd; i += 128) g[i] *= inv;
}

// ---------------------------------------------------------------------
// 9. out[b,c,i] = sum_j v1[b,c,j]*att[b,i,j]  (K=2000)
//    Wave: 16 rows x 80 cols. grid (25,8,8), block 32
// ---------------------------------------------------------------------
__global__ void attnout_kernel(const float* __restrict__ v1, const float* __restrict__ att,
                               float* __restrict__ out) {
  const int NB = 5;
  int nt = blockIdx.x, mt = blockIdx.y, b = blockIdx.z;
  int lane = threadIdx.x, l15 = lane & 15, hk = lane >> 4;
  const float* vb = v1 + (size_t)b * Cd * Nd;
  const float* ab = att + (size_t)b * Nd * Nd;
  int rowA = mt * 16 + l15;
  int col0 = nt * 16 * NB + l15;
  v8f acc[NB];
#pragma unroll
  for (int t = 0; t < NB; t++) acc[t] = (v8f){};
  for (int k = 0; k < Nd; k += 4) {
    int k0 = k + hk * 2;
    v2f a;
    a.x = vb[(size_t)rowA * Nd + k0];
    a.y = vb[(size_t)rowA * Nd + k0 + 1];
#pragma unroll
    for (int t = 0; t < NB; t++) {
      int colB = col0 + t * 16;
      v2f bb;
      bb.x = ab[(size_t)colB * Nd + k0];
      bb.y = ab[(size_t)colB * Nd + k0 + 1];
      acc[t] = wmma4(a, bb, acc[t]);
    }
  }
  float* ob = out + (size_t)b * Cd * Nd;
  int rowbase = mt * 16 + hk * 8;
#pragma unroll
  for (int t = 0; t < NB; t++) {
    AccU u; u.v = acc[t];
    int colB = col0 + t * 16;
    for (int r = 0; r < 8; r++) ob[(size_t)(rowbase + r) * Nd + colB] = u.f[r];
  }
}

// ---------------------------------------------------------------------
// 10. proj[b,c,e] = sum_n f[b,c,n]*eig[b,n,e]   grid (2,8,8)
// ---------------------------------------------------------------------
__global__ void spec_proj_kernel(const float* __restrict__ f, const float* __restrict__ eig,
                                 float* __restrict__ proj) {
  int nt = blockIdx.x, mt = blockIdx.y, b = blockIdx.z;
  int lane = threadIdx.x, l15 = lane & 15, hk = lane >> 4;
  const float* fb = f + (size_t)b * Cd * Nd;
  const float* eb = eig + (size_t)b * Nd * Ed;
  int rowA = mt * 16 + l15, colB = nt * 16 + l15;
  v8f acc = {};
  for (int k = 0; k < Nd; k += 4) {
    int k0 = k + hk * 2;
    v2f a, bb;
    a.x = fb[(size_t)rowA * Nd + k0]; a.y = fb[(size_t)rowA * Nd + k0 + 1];
    bb.x = eb[(size_t)k0 * Ed + colB]; bb.y = eb[(size_t)(k0 + 1) * Ed + colB];
    acc = wmma4(a, bb, acc);
  }
  AccU u; u.v = acc;
  float* pb = proj + (size_t)b * Cd * Ed;
  int rowbase = mt * 16 + hk * 8;
  for (int r = 0; r < 8; r++) pb[(rowbase + r) * Ed + colB] = u.f[r];
}

// ---------------------------------------------------------------------
// 11. corr[b,c,n] = sum_e (proj[b,c,e]*spec[e]) * eig[b,n,e]
//     Wave: 16 rows x 80 cols. grid (25,8,8), block 32
// ---------------------------------------------------------------------
__global__ void spec_recon_kernel(const float* __restrict__ proj, const float* __restrict__ spec,
                                  const float* __restrict__ eig, float* __restrict__ corr) {
  const int NB = 5;
  int nt = blockIdx.x, mt = blockIdx.y, b = blockIdx.z;
  int lane = threadIdx.x, l15 = lane & 15, hk = lane >> 4;
  const float* pb = proj + (size_t)b * Cd * Ed;
  const float* eb = eig + (size_t)b * Nd * Ed;
  int rowA = mt * 16 + l15;
  int col0 = nt * 16 * NB + l15;
  v8f acc[NB];
#pragma unroll
  for (int t = 0; t < NB; t++) acc[t] = (v8f){};
  for (int k = 0; k < Ed; k += 4) {
    int k0 = k + hk * 2;
    v2f a;
    a.x = pb[rowA * Ed + k0] * spec[k0];
    a.y = pb[rowA * Ed + k0 + 1] * spec[k0 + 1];
#pragma unroll
    for (int t = 0; t < NB; t++) {
      int colB = col0 + t * 16;
      v2f bb;
      bb.x = eb[(size_t)colB * Ed + k0];
      bb.y = eb[(size_t)colB * Ed + k0 + 1];
      acc[t] = wmma4(a, bb, acc[t]);
    }
  }
  float* cb = corr + (size_t)b * Cd * Nd;
  int rowbase = mt * 16 + hk * 8;
#pragma unroll
  for (int t = 0; t < NB; t++) {
    AccU u; u.v = acc[t];
    int colB = col0 + t * 16;
    for (int r = 0; r < 8; r++) cb[(size_t)(rowbase + r) * Nd + colB] = u.f[r];
  }
}

// ---------------------------------------------------------------------
// 12. alpha[b,n] = sigmoid(fc_b + sum_c fcw[c]*corr + fcw[C+c]*(corr-f))
// ---------------------------------------------------------------------
__global__ void alpha_kernel(const float* __restrict__ corr, const float* __restrict__ f,
                             const float* __restrict__ fcw, const float* __restrict__ fcb,
                             float* __restrict__ alpha) {
  int t = blockIdx.x * blockDim.x + threadIdx.x;
  if (t >= Bd * Nd) return;
  int b = t / Nd, n = t % Nd;
  const float* cb = corr + (size_t)b * Cd * Nd + n;
  const float* fb = f + (size_t)b * Cd * Nd + n;
  float s = fcb[0];
  for (int c = 0; c < Cd; c++) {
    float cv = cb[(size_t)c * Nd];
    float fv = fb[(size_t)c * Nd];
    s += fcw[c] * cv + fcw[Cd + c] * (cv - fv);
  }
  alpha[t] = 1.f / (1.f + __expf(-s));
}

// 13. hin = corr - alpha*f
__global__ void hin_kernel(const float* __restrict__ corr, const float* __restrict__ f,
                           const float* __restrict__ alpha, float* __restrict__ hin) {
  int t = blockIdx.x * blockDim.x + threadIdx.x;
  if (t >= Bd * Cd * Nd) return;
  int n = t % Nd;
  int b = t / (Cd * Nd);
  hin[t] = corr[t] - alpha[b * Nd + n] * f[t];
}

// =====================================================================
// Host launcher
// =====================================================================
extern "C" void kernel_launch(void* const* d_in, const int* in_sizes, int n_in,
                              void* d_out, int out_size, void* d_ws, size_t ws_size,
                              hipStream_t stream) {
  const float* feats = (const float*)d_in[0];     // (8,128,2000,1)
  const float* eig   = (const float*)d_in[1];     // (8,2000,32)
  // d_in[2] = eig_val (unused by reference forward)
  // params flattened as jax pytree (dict keys sorted, recursive):
  const float* af_b1   = (const float*)d_in[3];
  const float* af_b2   = (const float*)d_in[4];
  const float* af_bn1b = (const float*)d_in[5];
  const float* af_bn1g = (const float*)d_in[6];
  const float* af_bn2b = (const float*)d_in[7];
  const float* af_bn2g = (const float*)d_in[8];
  const float* af_fcb  = (const float*)d_in[9];
  const float* af_fcw  = (const float*)d_in[10];
  const float* af_w1   = (const float*)d_in[11];
  const float* af_w2   = (const float*)d_in[12];
  const float* cn0_b1 = (const float*)d_in[13], *cn0_b2 = (const float*)d_in[14];
  const float* cn0_bn1b = (const float*)d_in[15], *cn0_bn1g = (const float*)d_in[16];
  const float* cn0_bn2b = (const float*)d_in[17], *cn0_bn2g = (const float*)d_in[18];
  const float* cn0_w1 = (const float*)d_in[19], *cn0_w2 = (const float*)d_in[20];
  const float* cn1_b1 = (const float*)d_in[21], *cn1_b2 = (const float*)d_in[22];
  const float* cn1_bn1b = (const float*)d_in[23], *cn1_bn1g = (const float*)d_in[24];
  const float* cn1_bn2b = (const float*)d_in[25], *cn1_bn2g = (const float*)d_in[26];
  const float* cn1_w1 = (const float*)d_in[27], *cn1_w2 = (const float*)d_in[28];
  const float* cn2_b1 = (const float*)d_in[29], *cn2_b2 = (const float*)d_in[30];
  const float* cn2_bn1b = (const float*)d_in[31], *cn2_bn1g = (const float*)d_in[32];
  const float* cn2_bn2b = (const float*)d_in[33], *cn2_bn2g = (const float*)d_in[34];
  const float* cn2_w1 = (const float*)d_in[35], *cn2_w2 = (const float*)d_in[36];
  const float* spectrum = (const float*)d_in[37];
  const float* t0_a1b  = (const float*)d_in[38];
  const float* t0_a1bb = (const float*)d_in[39];
  const float* t0_a1g  = (const float*)d_in[40];
  const float* t0_a1w  = (const float*)d_in[41];
  const float* t0_gamma = (const float*)d_in[42];
  const float* t0_kb = (const float*)d_in[43], *t0_kw = (const float*)d_in[44];
  const float* t0_qb = (const float*)d_in[45], *t0_qw = (const float*)d_in[46];
  const float* t0_vb = (const float*)d_in[47], *t0_vw = (const float*)d_in[48];
  const float* t1_cb = (const float*)d_in[49], *t1_cbb = (const float*)d_in[50];
  const float* t1_cg = (const float*)d_in[51], *t1_cw = (const float*)d_in[52];
  const float* t1_gamma = (const float*)d_in[53];
  const float* t1_kb = (const float*)d_in[54], *t1_kbb = (const float*)d_in[55];
  const float* t1_kg = (const float*)d_in[56], *t1_kw = (const float*)d_in[57];
  const float* t1_qb = (const float*)d_in[58], *t1_qbb = (const float*)d_in[59];
  const float* t1_qg = (const float*)d_in[60], *t1_qw = (const float*)d_in[61];
  const float* t1_vb = (const float*)d_in[62], *t1_vbb = (const float*)d_in[63];
  const float* t1_vg = (const float*)d_in[64], *t1_vw = (const float*)d_in[65];
  (void)n_in; (void)in_sizes; (void)out_size; (void)ws_size;

  // ---- scratch layout (floats) : ~299 MB total ----
  float* ws = (float*)d_ws;
  size_t o = 0;
  float* G    = ws + o; o += (size_t)Bd * Nd * Nd;      // 32,000,000 (dist / scores)
  float* hpre = ws + o; o += (size_t)Bd * Cd * NK;      // 18,432,000
  const size_t BCN = (size_t)Bd * Cd * Nd;              // 2,048,000
  float* xTbuf     = ws + o; o += BCN;
  float* localfeat = ws + o; o += BCN;
  float* tmpA      = ws + o; o += BCN;
  float* fbuf      = ws + o; o += BCN;
  float* v1buf     = ws + o; o += BCN;
  float* attout    = ws + o; o += BCN;
  float* gout      = ws + o; o += BCN;
  float* f2buf     = ws + o; o += BCN;
  float* corr      = ws + o; o += BCN;
  float* hinb      = ws + o; o += BCN;
  float* aggb      = ws + o; o += BCN;
  float* q1buf = ws + o; o += (size_t)Bd * Ed * Nd;     // 512,000
  float* k1buf = ws + o; o += (size_t)Bd * Ed * Nd;
  float* proj  = ws + o; o += (size_t)Bd * Cd * Ed;
  float* xxbuf = ws + o; o += (size_t)Bd * Nd;
  float* alphab = ws + o; o += (size_t)Bd * Nd;
  float* statm = ws + o; o += (size_t)Bd * Cd;
  float* stati = ws + o; o += (size_t)Bd * Cd;
  int* idxbuf = (int*)(ws + o); o += (size_t)Bd * Nd * Kn;
  float* outp = (float*)d_out;

  dim3 w32(32);
  const int NB = 5;
  const int NT5 = Nd / (16 * NB);   // 25
  const int MT = Cd / 16;           // 8
  const int JT5 = NK / (16 * NB);   // 225

  // ---- Stage A: KNN ----
  xx_kernel<<<(Bd * Nd + 255) / 256, 256, 0, stream>>>(feats, xxbuf);
  transpose_kernel<<<(Bd * Cd * Nd + 255) / 256, 256, 0, stream>>>(feats, xTbuf);
  knn_dist_kernel<<<dim3(NT5, Nd / 16, Bd), w32, 0, stream>>>(xTbuf, xxbuf, G);
  topk_kernel<<<(Bd * Nd + 127) / 128, 128, 0, stream>>>(G, idxbuf);

  // ---- Stage B: local transformer ----
  local_h_kernel<<<dim3(JT5, MT, Bd), w32, 0, stream>>>(feats, idxbuf, t0_a1w, t0_a1b, hpre);
  stats_kernel<<<Bd * Cd, 256, 0, stream>>>(hpre, statm, stati, NK, EPS_CN);
  local_attn_kernel<<<dim3(Nd, Bd), 256, 0, stream>>>(
      feats, hpre, statm, stati, t0_a1g, t0_a1bb,
      t0_qw, t0_qb, t0_kw, t0_kb, t0_vw, t0_vb, t0_gamma, localfeat);

  // ---- Stage C: PointCN cn0 (localfeat -> fbuf) ----
  stats_kernel<<<Bd * Cd, 256, 0, stream>>>(localfeat, statm, stati, Nd, EPS_CN);
  gemm_kernel<1, 0, NB><<<dim3(NT5, MT, Bd), w32, 0, stream>>>(
      cn0_w1, cn0_b1, localfeat, Cd, Nd, statm, stati, cn0_bn1g, cn0_bn1b,
      nullptr, nullptr, nullptr, nullptr, tmpA);
  stats_kernel<<<Bd * Cd, 256, 0, stream>>>(tmpA, statm, stati, Nd, EPS_CN);
  gemm_kernel<1, 2, NB><<<dim3(NT5, MT, Bd), w32, 0, stream>>>(
      cn0_w2, cn0_b2, tmpA, Cd, Nd, statm, stati, cn0_bn2g, cn0_bn2b,
      nullptr, nullptr, localfeat, nullptr, fbuf);

  // ---- Stage D: global transformer (fbuf -> gout) ----
  gemm_kernel<0, 1, NB><<<dim3(NT5, Ed / 16, Bd), w32, 0, stream>>>(
      t1_qw, t1_qb, fbuf, Cd, Nd, nullptr, nullptr, nullptr, nullptr,
      t1_qg, t1_qbb, nullptr, nullptr, q1buf);
  gemm_kernel<0, 1, NB><<<dim3(NT5, Ed / 16, Bd), w32, 0, stream>>>(
      t1_kw, t1_kb, fbuf, Cd, Nd, nullptr, nullptr, nullptr, nullptr,
      t1_kg, t1_kbb, nullptr, nullptr, k1buf);
  gemm_kernel<0, 1, NB><<<dim3(NT5, MT, Bd), w32, 0, stream>>>(
      t1_vw, t1_vb, fbuf, Cd, Nd, nullptr, nullptr, nullptr, nullptr,
      t1_vg, t1_vbb, nullptr, nullptr, v1buf);
  scores_kernel<<<dim3(NT5, Nd / 16, Bd), w32, 0, stream>>>(q1buf, k1buf, G);
  softmax_kernel<<<Bd * Nd, 128, 0, stream>>>(G);
  attnout_kernel<<<dim3(NT5, MT, Bd), w32, 0, stream>>>(v1buf, G, attout);
  gemm_kernel<0, 3, NB><<<dim3(NT5, MT, Bd), w32, 0, stream>>>(
      t1_cw, t1_cb, attout, Cd, Nd, nullptr, nullptr, nullptr, nullptr,
      t1_cg, t1_cbb, fbuf, t1_gamma, gout);

  // ---- Stage E: PointCN cn1 (gout -> f2buf) ----
  stats_kernel<<<Bd * Cd, 256, 0, stream>>>(gout, statm, stati, Nd, EPS_CN);
  gemm_kernel<1, 0, NB><<<dim3(NT5, MT, Bd), w32, 0, stream>>>(
      cn1_w1, cn1_b1, gout, Cd, Nd, statm, stati, cn1_bn1g, cn1_bn1b,
      nullptr, nullptr, nullptr, nullptr, tmpA);
  stats_kernel<<<Bd * Cd, 256, 0, stream>>>(tmpA, statm, stati, Nd, EPS_CN);
  gemm_kernel<1, 2, NB><<<dim3(NT5, MT, Bd), w32, 0, stream>>>(
      cn1_w2, cn1_b2, tmpA, Cd, Nd, statm, stati, cn1_bn2g, cn1_bn2b,
      nullptr, nullptr, gout, nullptr, f2buf);

  // ---- Stage F: spectral GCN ----
  spec_proj_kernel<<<dim3(Ed / 16, MT, Bd), w32, 0, stream>>>(f2buf, eig, proj);
  spec_recon_kernel<<<dim3(NT5, MT, Bd), w32, 0, stream>>>(proj, spectrum, eig, corr);

  // ---- Stage G: AFAM ----
  alpha_kernel<<<(Bd * Nd + 255) / 256, 256, 0, stream>>>(corr, f2buf, af_fcw, af_fcb, alphab);
  hin_kernel<<<(Bd * Cd * Nd + 255) / 256, 256, 0, stream>>>(corr, f2buf, alphab, hinb);
  stats_kernel<<<Bd * Cd, 256, 0, stream>>>(hinb, statm, stati, Nd, EPS_AF);
  gemm_kernel<1, 0, NB><<<dim3(NT5, MT, Bd), w32, 0, stream>>>(
      af_w1, af_b1, hinb, Cd, Nd, statm, stati, af_bn1g, af_bn1b,
      nullptr, nullptr, nullptr, nullptr, tmpA);
  stats_kernel<<<Bd * Cd, 256, 0, stream>>>(tmpA, statm, stati, Nd, EPS_AF);
  gemm_kernel<1, 0, NB><<<dim3(NT5, MT, Bd), w32, 0, stream>>>(
      af_w2, af_b2, tmpA, Cd, Nd, statm, stati, af_bn2g, af_bn2b,
      nullptr, nullptr, nullptr, nullptr, aggb);

  // ---- Stage H: PointCN cn2 (aggb -> d_out) ----
  stats_kernel<<<Bd * Cd, 256, 0, stream>>>(aggb, statm, stati, Nd, EPS_CN);
  gemm_kernel<1, 0, NB><<<dim3(NT5, MT, Bd), w32, 0, stream>>>(
      cn2_w1, cn2_b1, aggb, Cd, Nd, statm, stati, cn2_bn1g, cn2_bn1b,
      nullptr, nullptr, nullptr, nullptr, tmpA);
  stats_kernel<<<Bd * Cd, 256, 0, stream>>>(tmpA, statm, stati, Nd, EPS_CN);
  gemm_kernel<1, 2, NB><<<dim3(NT5, MT, Bd), w32, 0, stream>>>(
      cn2_w2, cn2_b2, tmpA, Cd, Nd, statm, stati, cn2_bn2g, cn2_bn2b,
      nullptr, nullptr, aggb, nullptr, outp);
}